// ChamferLoss_47493748359773
// MI455X (gfx1250) — compile-verified
//
#include <hip/hip_runtime.h>
#include <math.h>

typedef __attribute__((ext_vector_type(2))) float v2f;
typedef __attribute__((ext_vector_type(4))) float v4f;
typedef __attribute__((ext_vector_type(8))) float v8f;

#define NPTS 4096
#define NB 8
#define ROWS_PER_WG 128
#define COL_TILES (NPTS / 16)
#define FLT_MAX_BITS 0x7F7FFFFF

static __device__ __forceinline__ int imin2(int a, int b) { return a < b ? a : b; }

// ---------------------------------------------------------------------------
// Phase 0: init the global column-min scratch to +FLT_MAX (bit pattern).
// ---------------------------------------------------------------------------
__global__ void chamfer_init_kernel(int* __restrict__ minB2) {
    int i = blockIdx.x * blockDim.x + threadIdx.x;
    if (i < NB * NPTS) minB2[i] = FLT_MAX_BITS;
}

// ---------------------------------------------------------------------------
// Phase 1: tiled all-pairs squared distances via V_WMMA_F32_16X16X4_F32.
// One workgroup = (batch b, 128-row block of set1), sweeps all 4096 cols.
// Each of the 8 waves owns one 16-row tile; one WMMA per 16x16 tile.
//   A row m = [-2x0, -2x1, -2x2, 1]   (K padded 3 -> 4)
//   B col n = [y0, y1, y2, ||y||^2]   (precomputed once into LDS as float4)
//   C[m][n] = ||x_m||^2   ==>   D = squared-distance tile
// Min tracking is integer-min on f32 bit patterns (order-preserving for
// non-negatives; stray negative d^2 sorts below all and clamps to 0 at
// readout, which equals max(min(d^2), 0) since clamp is monotone).
// ---------------------------------------------------------------------------
__global__ __launch_bounds__(256)
void chamfer_tiles_kernel(const float* __restrict__ set1,
                          const float* __restrict__ set2,
                          int* __restrict__ minA2,   // [NB*NPTS] row-min d^2 bits
                          int* __restrict__ minB2) { // [NB*NPTS] col-min d^2 bits
    __shared__ v4f byLDS[NPTS];   // 64 KB: padded B panel (y0,y1,y2,||y||^2)
    __shared__ int colmin[NPTS];  // 16 KB: per-column running min (bits)

    const int b       = blockIdx.y;
    const int rowBase = blockIdx.x * ROWS_PER_WG;
    const int tid     = threadIdx.x;
    const int lane    = tid & 31;   // wave32
    const int wave    = tid >> 5;
    const int m       = lane & 15;
    const int hi      = lane >> 4;  // 0: K=0,1 / M=r ; 1: K=2,3 / M=r+8

    const float* s1b = set1 + (size_t)b * NPTS * 3;
    const float* s2b = set2 + (size_t)b * NPTS * 3;

    // ---- build the padded B panel + init column mins (once per WG) ----
    for (int j = tid; j < NPTS; j += 256) {
        const float* p = s2b + (size_t)j * 3;
        float y0 = p[0], y1 = p[1], y2 = p[2];
        v4f q;
        q.x = y0; q.y = y1; q.z = y2;
        q.w = y0 * y0 + y1 * y1 + y2 * y2;
        byLDS[j]  = q;
        colmin[j] = FLT_MAX_BITS;
    }
    __syncthreads();

    const int r0 = rowBase + wave * 16;

    // ---- A fragment (16x4 f32): lanes<16 -> K0,K1 ; lanes>=16 -> K2,K3 ----
    const float* pa = s1b + (size_t)(r0 + m) * 3;
    float ax0 = pa[0], ax1 = pa[1], ax2 = pa[2];
    v2f afrag;
    afrag.x = hi ? (-2.0f * ax2) : (-2.0f * ax0);
    afrag.y = hi ? 1.0f          : (-2.0f * ax1);

    // ---- C fragment: row norms ||x||^2, VGPR r -> row r0 + r + 8*hi ----
    v8f cinit;
#pragma unroll
    for (int r = 0; r < 8; ++r) {
        const float* pr = s1b + (size_t)(r0 + r + 8 * hi) * 3;
        float q0 = pr[0], q1 = pr[1], q2 = pr[2];
        cinit[r] = q0 * q0 + q1 * q1 + q2 * q2;
    }

    int rmin[8];
#pragma unroll
    for (int r = 0; r < 8; ++r) rmin[r] = FLT_MAX_BITS;

    // LDS viewed as pairs of floats: element (point)*2 + hi selects either
    // (y0,y1) or (y2,||y||^2) -> one conflict-free ds_load_b64 per tile.
    const v2f* byPairs = (const v2f*)byLDS;
    const int  pairIdx = m * 2 + hi;

#pragma unroll 2
    for (int ct = 0; ct < COL_TILES; ++ct) {
        const int c0 = ct * 16;

        // ---- B fragment straight from LDS (no VALU prep needed) ----
        v2f bfrag = byPairs[c0 * 2 + pairIdx];

        // D = A*B + C : 16x16 tile of squared distances
        v8f d = __builtin_amdgcn_wmma_f32_16x16x4_f32(
            /*neg_a=*/false, afrag, /*neg_b=*/false, bfrag,
            /*c_mod=*/(short)0, cinit, /*reuse_a=*/false, /*reuse_b=*/false);

        // integer-min on bit patterns: no NaN-canonicalize VALU noise
        int di[8];
#pragma unroll
        for (int r = 0; r < 8; ++r) di[r] = __float_as_int(d[r]);
#pragma unroll
        for (int r = 0; r < 8; ++r) rmin[r] = imin2(rmin[r], di[r]);

        // column min for this tile (min3-friendly tree)
        int cm = imin2(imin2(imin2(di[0], di[1]), di[2]),
                       imin2(imin2(di[3], di[4]),
                             imin2(imin2(di[5], di[6]), di[7])));
        // all 32 lanes participate: lanes L and L+16 cover rows 0-7 / 8-15
        // of the same column; LDS atomic unit merges the 2-way conflict.
        atomicMin(&colmin[c0 + m], cm);
    }

    // ---- finalize row mins: reduce across the 16 lanes of each half-group ----
#pragma unroll
    for (int r = 0; r < 8; ++r) {
        int v = rmin[r];
        v = imin2(v, __shfl_xor(v, 1, 32));
        v = imin2(v, __shfl_xor(v, 2, 32));
        v = imin2(v, __shfl_xor(v, 4, 32));
        v = imin2(v, __shfl_xor(v, 8, 32));
        rmin[r] = v;
    }
    if (m == 0) {  // lanes 0 and 16 each own 8 rows; rows unique per wave
#pragma unroll
        for (int r = 0; r < 8; ++r) {
            int row = r0 + r + 8 * hi;
            minA2[b * NPTS + row] = rmin[r];
        }
    }

    // ---- fold LDS column mins into the global scratch ----
    __syncthreads();
    for (int j = tid; j < NPTS; j += 256) {
        atomicMin(&minB2[b * NPTS + j], colmin[j]);
    }
}

// ---------------------------------------------------------------------------
// Phase 2: sqrt + mean reduction to the scalar output.
//   out = (mean(sqrt(max(minA2,0))) + mean(sqrt(max(minB2,0)))) / NPTS
// ---------------------------------------------------------------------------
__global__ __launch_bounds__(256)
void chamfer_reduce_kernel(const int* __restrict__ minA2,
                           const int* __restrict__ minB2,
                           float* __restrict__ out) {
    __shared__ float red[256];
    float s = 0.0f;
    for (int i = threadIdx.x; i < NB * NPTS; i += 256) {
        s += sqrtf(fmaxf(__int_as_float(minA2[i]), 0.0f));
        s += sqrtf(fmaxf(__int_as_float(minB2[i]), 0.0f));
    }
    red[threadIdx.x] = s;
    __syncthreads();
    for (int off = 128; off > 0; off >>= 1) {
        if (threadIdx.x < off) red[threadIdx.x] += red[threadIdx.x + off];
        __syncthreads();
    }
    if (threadIdx.x == 0) {
        out[0] = red[0] / (float)(NB * NPTS) / (float)NPTS;
    }
}

// ---------------------------------------------------------------------------
extern "C" void kernel_launch(void* const* d_in, const int* in_sizes, int n_in,
                              void* d_out, int out_size, void* d_ws, size_t ws_size,
                              hipStream_t stream) {
    (void)in_sizes; (void)n_in; (void)out_size; (void)ws_size;

    const float* set1 = (const float*)d_in[0];  // [8, 4096, 3] f32
    const float* set2 = (const float*)d_in[1];  // [8, 4096, 3] f32

    int* minA2 = (int*)d_ws;                                   // 32768 i32
    int* minB2 = (int*)((char*)d_ws + (size_t)NB * NPTS * 4);  // 32768 i32

    chamfer_init_kernel<<<(NB * NPTS + 255) / 256, 256, 0, stream>>>(minB2);

    dim3 grid(NPTS / ROWS_PER_WG, NB);  // (32, 8) workgroups
    chamfer_tiles_kernel<<<grid, 256, 0, stream>>>(set1, set2, minA2, minB2);

    chamfer_reduce_kernel<<<1, 256, 0, stream>>>(minA2, minB2, (float*)d_out);
}